// EmbeddingBagCollection_13692355739720
// MI455X (gfx1250) — compile-verified
//
#include <hip/hip_runtime.h>
#include <hip/hip_bf16.h>

// EmbeddingBagCollection mean-pool, MI455X (gfx1250).
// Memory-bound random gather: 1.34 GB of 512B rows at 23.3 TB/s ~= 60us floor
// (less with L2 reuse: 410MB table set, ~3.3x re-reference, 192MB L2).
// Fetch path: CDNA5 async global->LDS loads (ASYNCcnt), 16-deep per-wave ring;
// consume loop is ds_load_b128 + v_add. Output stored non-temporally to keep
// L2 for the tables.

#define NUM_TABLES 8
#define DIM 128            // floats per row -> 512 bytes
#define ROW_BYTES (DIM * 4)
#define DEPTH 16           // async pipeline depth (rows in flight per wave)
#define SCAN_THREADS 1024

typedef float v4f __attribute__((ext_vector_type(4)));   // native clang vector

// ---------------- Kernel 1: exclusive prefix sum of lengths -> offsets ------
__global__ __launch_bounds__(SCAN_THREADS)
void ebc_scan_kernel(const int* __restrict__ lengths, int* __restrict__ offsets, int B) {
    __shared__ int s[SCAN_THREADS];
    const int tid = threadIdx.x;
    const int per = (B + SCAN_THREADS - 1) / SCAN_THREADS;
    const int begin = tid * per;
    const int end = (begin + per < B) ? (begin + per) : B;

    int lsum = 0;
    for (int i = begin; i < end; ++i) lsum += lengths[i];
    s[tid] = lsum;
    __syncthreads();

    // Hillis-Steele inclusive scan over the 1024 partial sums.
    for (int d = 1; d < SCAN_THREADS; d <<= 1) {
        int add = (tid >= d) ? s[tid - d] : 0;
        __syncthreads();
        s[tid] += add;
        __syncthreads();
    }

    int run = (tid == 0) ? 0 : s[tid - 1];
    for (int i = begin; i < end; ++i) {
        offsets[i] = run;
        run += lengths[i];
    }
}

// ---------------- Kernel 2: one block per bag, one wave per table -----------
__global__ __launch_bounds__(NUM_TABLES * 32)
void ebc_pool_kernel(const float* __restrict__ tables,
                     const int* __restrict__ values,
                     const int* __restrict__ offsets,
                     const int* __restrict__ lengths,
                     float* __restrict__ out,
                     int vocab) {
    // Per-wave private ring: DEPTH slots of one 512B row each (64KB total).
    __shared__ __align__(16) char smem[NUM_TABLES * DEPTH * ROW_BYTES];

    const int bag  = blockIdx.x;
    const int lane = threadIdx.x & 31;           // wave32
    const int w    = threadIdx.x >> 5;           // wave id == table id (uniform per wave)

    const int start = offsets[bag];              // uniform -> s_load
    const int len   = lengths[bag];              // uniform -> s_load

    const float* tbase = tables + (size_t)w * (size_t)vocab * (size_t)DIM; // wave-uniform SADDR
    const size_t waveOff = (size_t)w * DEPTH * ROW_BYTES;
    const unsigned ldsWave = (unsigned)(size_t)&smem[waveOff];  // low 32b of generic LDS addr
    const unsigned laneB = (unsigned)lane * 16u;

    // ---- prologue: fill the async pipeline ----
    const int nPro = (len < DEPTH) ? len : DEPTH;
    for (int p = 0; p < nPro; ++p) {
        const int idx = values[start + p];                 // uniform scalar load
        const unsigned la = ldsWave + (unsigned)p * (unsigned)ROW_BYTES + laneB;
        const unsigned ga = (unsigned)idx * (unsigned)ROW_BYTES + laneB;
        asm volatile("global_load_async_to_lds_b128 %0, %1, %2"
                     :: "v"(la), "v"(ga), "s"(tbase) : "memory");
    }

    float a0 = 0.f, a1 = 0.f, a2 = 0.f, a3 = 0.f;

    // ---- steady state: wait row j (in-order completion), consume, reissue ----
    int mainN = len - DEPTH;
    if (mainN < 0) mainN = 0;
    for (int j = 0; j < mainN; ++j) {
        // issued = j + DEPTH; asynccnt<=DEPTH-1 => row j has landed in LDS.
        asm volatile("s_wait_asynccnt 0xf" ::: "memory");
        const v4f v = *(const v4f*)&smem[waveOff +
                             (size_t)(j & (DEPTH - 1)) * ROW_BYTES + (size_t)lane * 16];
        a0 += v.x; a1 += v.y; a2 += v.z; a3 += v.w;

        const int idx = values[start + j + DEPTH];         // uniform scalar load
        const unsigned la = ldsWave + (unsigned)(j & (DEPTH - 1)) * (unsigned)ROW_BYTES + laneB;
        const unsigned ga = (unsigned)idx * (unsigned)ROW_BYTES + laneB;
        // Slot reuse: make sure our ds_load of this slot retired before the
        // async engine can overwrite it, then refill.
        asm volatile("s_wait_dscnt 0x0\n\t"
                     "global_load_async_to_lds_b128 %0, %1, %2"
                     :: "v"(la), "v"(ga), "s"(tbase) : "memory");
    }

    // ---- drain ----
    asm volatile("s_wait_asynccnt 0x0" ::: "memory");
    for (int j = mainN; j < len; ++j) {
        const v4f v = *(const v4f*)&smem[waveOff +
                             (size_t)(j & (DEPTH - 1)) * ROW_BYTES + (size_t)lane * 16];
        a0 += v.x; a1 += v.y; a2 += v.z; a3 += v.w;
    }

    const float scale = 1.0f / (float)((len > 0) ? len : 1);
    v4f r;
    r.x = a0 * scale; r.y = a1 * scale; r.z = a2 * scale; r.w = a3 * scale;
    // out[bag, w*128 + lane*4 .. +3]; 512B coalesced per wave; NT to spare L2.
    v4f* dst = (v4f*)&out[(size_t)bag * (NUM_TABLES * DIM) + (size_t)w * DIM + (size_t)lane * 4];
    __builtin_nontemporal_store(r, dst);
}

// ---------------- launcher ---------------------------------------------------
extern "C" void kernel_launch(void* const* d_in, const int* in_sizes, int n_in,
                              void* d_out, int out_size, void* d_ws, size_t ws_size,
                              hipStream_t stream) {
    const float* tables  = (const float*)d_in[0];   // (8, 100000, 128) f32
    const int*   values  = (const int*)d_in[1];     // (L,) i32
    const int*   lengths = (const int*)d_in[2];     // (B,) i32
    const int B = in_sizes[2];
    const int vocab = (int)((size_t)in_sizes[0] / (size_t)(NUM_TABLES * DIM));

    int* offsets = (int*)d_ws;                      // B ints of scratch

    ebc_scan_kernel<<<1, SCAN_THREADS, 0, stream>>>(lengths, offsets, B);
    ebc_pool_kernel<<<B, NUM_TABLES * 32, 0, stream>>>(
        tables, values, offsets, lengths, (float*)d_out, vocab);
}